// GCN_15556371546547
// MI455X (gfx1250) — compile-verified
//
#include <hip/hip_runtime.h>
#include <hip/hip_bf16.h>

typedef __attribute__((ext_vector_type(2))) float v2f;
typedef __attribute__((ext_vector_type(8))) float v8f;

// ---------------------------------------------------------------------------
// Degree / normalization kernels
// ---------------------------------------------------------------------------
__global__ void k_init_deg(float* deg, int n) {
    int i = blockIdx.x * blockDim.x + threadIdx.x;
    if (i < n) deg[i] = 1.0f;               // self-loop contributes 1
}

__global__ void k_accum_deg(const int* __restrict__ dst, float* deg, int e) {
    int i = blockIdx.x * blockDim.x + threadIdx.x;
    if (i < e) atomicAdd(&deg[dst[i]], 1.0f);
}

__global__ void k_rsqrt(float* deg, int n) {
    int i = blockIdx.x * blockDim.x + threadIdx.x;
    if (i < n) deg[i] = rsqrtf(deg[i]);     // deg >= 1 always
}

// ---------------------------------------------------------------------------
// fp32 WMMA GEMM:  Y[nrows x 64] = X[nrows x K] * W[K x 64]
// One wave -> one 16x64 tile. W staged in LDS per workgroup (8 waves).
// Uses V_WMMA_F32_16X16X4_F32 (exact fp32 math).
// ---------------------------------------------------------------------------
template <int K>
__global__ __launch_bounds__(256) void k_gemm_wmma(const float* __restrict__ X,
                                                   const float* __restrict__ W,
                                                   float* __restrict__ Y,
                                                   int nrows) {
    __shared__ float sW[K * 64];
    for (int i = threadIdx.x; i < K * 64; i += blockDim.x) sW[i] = W[i];
    __syncthreads();

    const int wave = threadIdx.x >> 5;
    const int lane = threadIdx.x & 31;
    const int row0 = (blockIdx.x * 8 + wave) * 16;
    if (row0 >= nrows) return;              // wave-uniform: EXEC stays full for WMMA

    const int half = lane >> 4;             // 0: K=k0,k0+1   1: K=k0+2,k0+3
    const int l15  = lane & 15;

    v8f acc[4] = {v8f{}, v8f{}, v8f{}, v8f{}};

    for (int k0 = 0; k0 < K; k0 += 4) {
        const int ka = k0 + 2 * half;
        // A 16x4: lane l15 -> row (row0+l15), elements ka, ka+1
        const float* ap = X + (size_t)(row0 + l15) * K + ka;
        v2f a;
        a.x = ap[0];
        a.y = ap[1];
#pragma unroll
        for (int t = 0; t < 4; ++t) {
            // B 4x16: lane l15 -> col (t*16+l15), rows ka, ka+1
            const float* bp = sW + ka * 64 + t * 16 + l15;
            v2f b;
            b.x = bp[0];
            b.y = bp[64];
            acc[t] = __builtin_amdgcn_wmma_f32_16x16x4_f32(
                /*neg_a=*/false, a, /*neg_b=*/false, b,
                /*c_mod=*/(short)0, acc[t], /*reuse_a=*/false, /*reuse_b=*/false);
        }
    }

    // D layout: VGPR r, lanes 0-15 -> M=r; lanes 16-31 -> M=8+r; N=l15
#pragma unroll
    for (int t = 0; t < 4; ++t) {
        float* yp = Y + (size_t)(row0 + 8 * half) * 64 + t * 16 + l15;
#pragma unroll
        for (int r = 0; r < 8; ++r) yp[(size_t)r * 64] = acc[t][r];
    }
}

// ---------------------------------------------------------------------------
// agg[i][j] = xw[i][j] * dis[i]^2 + bias[j]   (self-loop term + bias init)
// ---------------------------------------------------------------------------
__global__ void k_init_agg(const float* __restrict__ xw, const float* __restrict__ dis,
                           const float* __restrict__ bias, float* __restrict__ agg,
                           int n) {
    int idx = blockIdx.x * blockDim.x + threadIdx.x;
    if (idx >= n * 64) return;
    int i = idx >> 6;
    int j = idx & 63;
    float d = dis[i];
    agg[idx] = xw[idx] * d * d + bias[j];
}

// ---------------------------------------------------------------------------
// Edge scatter: agg[dst] += xw[src] * (dis[src]*dis[dst]); 16 lanes / edge.
// ---------------------------------------------------------------------------
__global__ void k_scatter(const int* __restrict__ src, const int* __restrict__ dst,
                          const float* __restrict__ dis, const float* __restrict__ xw,
                          float* __restrict__ agg, int nedges) {
    int idx = blockIdx.x * blockDim.x + threadIdx.x;
    int e = idx >> 4;
    int t = idx & 15;
    if (e >= nedges) return;
    int s = src[e];
    int d = dst[e];
    float norm = dis[s] * dis[d];
    const float4 v = *(const float4*)(xw + (size_t)s * 64 + t * 4);
    float* o = agg + (size_t)d * 64 + t * 4;
    atomicAdd(o + 0, v.x * norm);
    atomicAdd(o + 1, v.y * norm);
    atomicAdd(o + 2, v.z * norm);
    atomicAdd(o + 3, v.w * norm);
}

// ---------------------------------------------------------------------------
// SELU in place
// ---------------------------------------------------------------------------
__global__ void k_selu(float* h, int total) {
    int i = blockIdx.x * blockDim.x + threadIdx.x;
    if (i >= total) return;
    const float alpha = 1.6732632423543772f;
    const float scale = 1.0507009873554805f;
    float x = h[i];
    h[i] = scale * (x > 0.0f ? x : alpha * expm1f(x));
}

// ---------------------------------------------------------------------------
// Launch
// ---------------------------------------------------------------------------
extern "C" void kernel_launch(void* const* d_in, const int* in_sizes, int n_in,
                              void* d_out, int out_size, void* d_ws, size_t ws_size,
                              hipStream_t stream) {
    const float* x  = (const float*)d_in[0];   // [N,128]
    const int*   ei = (const int*)d_in[1];     // [2,E]
    const float* W1 = (const float*)d_in[2];   // [128,64]
    const float* b1 = (const float*)d_in[3];   // [64]
    const float* W2 = (const float*)d_in[4];   // [64,64]
    const float* b2 = (const float*)d_in[5];   // [64]
    float* out = (float*)d_out;                // [N,64]

    const int N = in_sizes[0] / 128;
    const int E = in_sizes[1] / 2;
    const int* srcp = ei;
    const int* dstp = ei + E;

    // Workspace carve-up (256B aligned)
    char* ws = (char*)d_ws;
    size_t off = 0;
    float* dis  = (float*)(ws + off); off += ((size_t)N * 4 + 255) & ~(size_t)255;
    float* bufA = (float*)(ws + off); off += ((size_t)N * 64 * 4 + 255) & ~(size_t)255;
    float* bufB = (float*)(ws + off); off += ((size_t)N * 64 * 4 + 255) & ~(size_t)255;
    (void)ws_size; (void)n_in; (void)out_size;

    const int TB = 256;
    const int gN   = (N + TB - 1) / TB;
    const int gE   = (E + TB - 1) / TB;
    const int gNF  = (N * 64 + TB - 1) / TB;
    const int gE16 = ((E * 16) + TB - 1) / TB;
    const int gGemm = (((N + 15) / 16) + 7) / 8;   // 8 waves (16 rows each) per block

    // 1) symmetric normalization
    k_init_deg<<<gN, TB, 0, stream>>>(dis, N);
    k_accum_deg<<<gE, TB, 0, stream>>>(dstp, dis, E);
    k_rsqrt<<<gN, TB, 0, stream>>>(dis, N);

    // 2) layer 1: xw = x @ W1  -> bufA ; agg -> bufB
    k_gemm_wmma<128><<<gGemm, TB, 0, stream>>>(x, W1, bufA, N);
    k_init_agg<<<gNF, TB, 0, stream>>>(bufA, dis, b1, bufB, N);
    k_scatter<<<gE16, TB, 0, stream>>>(srcp, dstp, dis, bufA, bufB, E);
    k_selu<<<gNF, TB, 0, stream>>>(bufB, N * 64);

    // 3) layer 2: hw = h @ W2 -> bufA ; agg -> out
    k_gemm_wmma<64><<<gGemm, TB, 0, stream>>>(bufB, W2, bufA, N);
    k_init_agg<<<gNF, TB, 0, stream>>>(bufA, dis, b2, out, N);
    k_scatter<<<gE16, TB, 0, stream>>>(srcp, dstp, dis, bufA, out, E);
}